// WDM3DLoss_82102594830923
// MI455X (gfx1250) — compile-verified
//
#include <hip/hip_runtime.h>
#include <hip/hip_bf16.h>
#include <stdint.h>

typedef __attribute__((ext_vector_type(2))) float v2f;
typedef __attribute__((ext_vector_type(8))) float v8f;

#define PI_F  3.14159265358979323846f
#define HPI_F 1.57079632679489661923f

__device__ __forceinline__ float fast_rcp(float x)  { return __builtin_amdgcn_rcpf(x); }
__device__ __forceinline__ float fast_sqrt(float x) { return __builtin_amdgcn_sqrtf(x); }

// One workgroup (8 wave32s, 256 threads) per box.
// Streaming: async global->LDS double buffering, b128/b64 granularity.
// Reduction: V_WMMA_F32_16X16X4_F32 per-wave (sum,count) reduction.
__global__ __launch_bounds__(256)
void wdm3d_box_kernel(const float* __restrict__ wl,
                      const float* __restrict__ Ry,
                      const float* __restrict__ points,
                      const float* __restrict__ density,
                      const float* __restrict__ center,
                      float* __restrict__ box_mean_out,
                      float* __restrict__ box_valid_out,
                      int P)
{
    const int n    = blockIdx.x;
    const int tid  = threadIdx.x;
    const int TILE = 512;                // points per tile, 2 per thread

    __shared__ float2 sm_pts[2][512];    // 8 KB
    __shared__ float  sm_den[2][512];    // 4 KB
    __shared__ float  sm_red[8][32];     // 1 KB

    // ---- per-box constants (redundant per thread; tiny, L2-resident) ----
    const float w  = wl[2*n + 0];
    const float l  = wl[2*n + 1];
    const float ry = Ry[n];
    const float c0 = center[2*n + 0];
    const float c1 = center[2*n + 1];

    const float th0 = atanf(w / l);
    const float len = fast_sqrt(w*w + l*l) * 0.5f;

    float angs[4] = { th0 + ry, PI_F - th0 + ry, PI_F + th0 + ry, -th0 + ry };
    float cxa[4], cya[4];
#pragma unroll
    for (int j = 0; j < 4; ++j) {
        cxa[j] = len * __cosf(angs[j]) + c0;   // v_cos_f32: args bounded
        cya[j] = len * __sinf(angs[j]) + c1;
    }

    float ry2 = (ry == HPI_F) ? (ry - 1e-4f) : ry;
    ry2 = (ry2 == 0.0f) ? (ry2 + 1e-4f) : ry2;
    const float k1 = __sinf(ry2) / __cosf(ry2);                     // tan(ry2)
    const float k2 = __sinf(ry2 + HPI_F) / __cosf(ry2 + HPI_F);     // tan(ry2+pi/2)

    const float b11 = cya[0] - k1 * cxa[0];
    const float b12 = cya[2] - k1 * cxa[2];
    const float b21 = cya[0] - k2 * cxa[0];
    const float b22 = cya[2] - k2 * cxa[2];

    const float ks[4] = { k1, k2, k1, k2 };
    const float bs[4] = { b11, b22, b12, b21 };

    float cxr[4], cyr[4];
#pragma unroll
    for (int j = 0; j < 4; ++j) {
        cxr[j] = rintf(cxa[j] * 1e4f);   // rint == round-half-even == jnp.round
        cyr[j] = rintf(cya[j] * 1e4f);
    }
    float lox[4], hix[4], loy[4], hiy[4];
#pragma unroll
    for (int j = 0; j < 4; ++j) {
        const int jn = (j + 1) & 3;
        lox[j] = fminf(cxr[j], cxr[jn]);
        hix[j] = fmaxf(cxr[j], cxr[jn]);
        loy[j] = fminf(cyr[j], cyr[jn]);
        hiy[j] = fmaxf(cyr[j], cyr[jn]);
    }

    // ---- streaming: async global -> LDS, double-buffered, 16B/lane ----
    const uint64_t ptsBase = (uint64_t)(uintptr_t)(points  + (size_t)n * (size_t)P * 2);
    const uint64_t denBase = (uint64_t)(uintptr_t)(density + (size_t)n * (size_t)P);
    const int nt    = (P + TILE - 1) / TILE;
    const uint32_t npair = (uint32_t)(P >> 1);

    auto stage = [&](int t, int b) {
        uint32_t pair = (uint32_t)(t * (TILE / 2) + tid);           // 2 points per lane
        uint32_t cp   = pair < npair ? pair : (npair - 1u);         // clamp tail
        uint32_t voffP = cp * 16u;                                  // 2 x float2
        uint32_t voffD = cp * 8u;                                   // 2 x float
        // low 32 bits of generic pointer == LDS byte offset (ISA 10.2 aperture map)
        uint32_t ldsP = (uint32_t)(uintptr_t)&sm_pts[b][2*tid];     // 16B aligned
        uint32_t ldsD = (uint32_t)(uintptr_t)&sm_den[b][2*tid];     // 8B aligned
        asm volatile("global_load_async_to_lds_b128 %0, %1, %2"
                     :: "v"(ldsP), "v"(voffP), "s"(ptsBase) : "memory");
        asm volatile("global_load_async_to_lds_b64 %0, %1, %2"
                     :: "v"(ldsD), "v"(voffD), "s"(denBase) : "memory");
    };

    float accS = 0.0f;   // sum of selected distances (masked)
    float accC = 0.0f;   // count of dis_in points

    stage(0, 0);
    for (int t = 0; t < nt; ++t) {
        if (t + 1 < nt) {
            stage(t + 1, (t + 1) & 1);
            asm volatile("s_wait_asynccnt 0x2" ::: "memory");  // tile t landed
        } else {
            asm volatile("s_wait_asynccnt 0x0" ::: "memory");
        }

        const int b = t & 1;
#pragma unroll
        for (int u = 0; u < 2; ++u) {
            const int idx = t * TILE + 2 * tid + u;
            if (idx < P) {
                float2 pt  = sm_pts[b][2*tid + u];
                float  dns = sm_den[b][2*tid + u];

                float px = (pt.x == 0.0f) ? 1e-4f : pt.x;
                float py = pt.y;
                float slope = py * fast_rcp(px);

                int   mcount = 0;
                float best   = __builtin_inff();
                float zdis   = 0.0f;
#pragma unroll
                for (int j = 0; j < 4; ++j) {
                    float denom = slope - ks[j];
                    float ix = bs[j] * fast_rcp(denom);
                    float iy = ix * slope;                 // == bs*slope/denom
                    float d  = fabsf(ix - px) + fabsf(iy - py);
                    float cn = fast_sqrt(ix*ix + iy*iy);
                    float irx = rintf(ix * 1e4f);
                    float iry = rintf(iy * 1e4f);
                    // branchless mask: bitwise combine the 4 compares
                    int m = ((int)(irx > lox[j]) & (int)(irx < hix[j])) |
                            ((int)(iry > loy[j]) & (int)(iry < hiy[j]));
                    mcount += m;
                    float cm = m ? cn : __builtin_inff();
                    if (j == 0) { best = cm; zdis = d; }           // unrolled: compile-time
                    else {
                        int take = (int)(cm < best);               // first-min tiebreak
                        best = take ? cm : best;
                        zdis = take ? d  : zdis;
                    }
                }
                // branchless masked accumulate
                float sel = (mcount == 2) ? 1.0f : 0.0f;
                accS += sel * (zdis * fast_rcp(dns));
                accC += sel;
            }
        }
    }

    // ---- wave-level reduction via WMMA f32 16x16x4 ----
    // A[lane<16]:  (M=lane,    K=0)=accS, (K=1)=accC
    // A[lane>=16]: (M=lane-16, K=2)=accS, (K=3)=accC
    // B col 0 selects even K (sums), col 1 selects odd K (counts).
    const int lane = tid & 31;
    const int wid  = tid >> 5;
    v2f a; a.x = accS; a.y = accC;
    v2f bm;
    bm.x = (lane == 0 || lane == 16) ? 1.0f : 0.0f;  // K=0 / K=2 rows -> col 0
    bm.y = (lane == 1 || lane == 17) ? 1.0f : 0.0f;  // K=1 / K=3 rows -> col 1
    v8f cacc = {};
    cacc = __builtin_amdgcn_wmma_f32_16x16x4_f32(false, a, false, bm,
                                                 (short)0, cacc, false, false);
    // lane L holds D[0..7][L%16] (L<16) or D[8..15][L-16]; fold the 8 rows:
    float tsum = cacc[0] + cacc[1] + cacc[2] + cacc[3]
               + cacc[4] + cacc[5] + cacc[6] + cacc[7];
    sm_red[wid][lane] = tsum;
    __syncthreads();

    if (tid == 0) {
        float S = 0.0f, C = 0.0f;
        for (int wv = 0; wv < 8; ++wv) {
            S += sm_red[wv][0] + sm_red[wv][16];   // column 0 full sum
            C += sm_red[wv][1] + sm_red[wv][17];   // column 1 full sum
        }
        bool  valid = (C >= 2.5f);                 // cnt >= 3 (C is an exact int)
        float mean  = S / fmaxf(C, 1.0f);
        box_mean_out[n]  = valid ? mean : 0.0f;
        box_valid_out[n] = valid ? 1.0f : 0.0f;
    }
}

__global__ __launch_bounds__(256)
void wdm3d_finalize_kernel(const float* __restrict__ wsm,
                           const float* __restrict__ wsv,
                           float* __restrict__ out, int N)
{
    __shared__ float s1[256];
    __shared__ float s2[256];
    const int tid = threadIdx.x;
    float a = 0.0f, c = 0.0f;
    for (int i = tid; i < N; i += 256) { a += wsm[i]; c += wsv[i]; }
    s1[tid] = a; s2[tid] = c;
    __syncthreads();
    for (int s = 128; s > 0; s >>= 1) {
        if (tid < s) { s1[tid] += s1[tid + s]; s2[tid] += s2[tid + s]; }
        __syncthreads();
    }
    if (tid == 0) out[0] = s1[0] / fmaxf(s2[0], 1.0f);
}

extern "C" void kernel_launch(void* const* d_in, const int* in_sizes, int n_in,
                              void* d_out, int out_size, void* d_ws, size_t ws_size,
                              hipStream_t stream) {
    const float* wl      = (const float*)d_in[0];
    const float* Ry      = (const float*)d_in[1];
    const float* points  = (const float*)d_in[2];
    const float* density = (const float*)d_in[3];
    const float* center  = (const float*)d_in[4];

    const int N = in_sizes[0] / 2;        // wl is (N,2)
    const int P = in_sizes[3] / N;        // density is (N,P)

    float* wsm = (float*)d_ws;            // per-box masked means
    float* wsv = wsm + N;                 // per-box valid flags

    wdm3d_box_kernel<<<N, 256, 0, stream>>>(wl, Ry, points, density, center,
                                            wsm, wsv, P);
    wdm3d_finalize_kernel<<<1, 256, 0, stream>>>(wsm, wsv, (float*)d_out, N);
}